// RankingLoss_34256659153340
// MI455X (gfx1250) — compile-verified
//
#include <hip/hip_runtime.h>
#include <hip/hip_bf16.h>

// ---------------------------------------------------------------------------
// Pairwise ranking hinge loss, N = 16384, MARGIN = 0.1
//   D[i,j] = margin - p_i + p_j   <- rank-2 bilinear => v_wmma_f32_16x16x4_f32
//   h      = take ? relu(D) : 0   <- v_cndmask + v_med3 (1-op ReLU)
//   cnt    = per-lane f32 accum   <- dual cndmask + v_pk_add (VALU-balanced;
//                                    ballot/popcount made SALU the bottleneck)
// LDS loads are software-pipelined one tile ahead to hide ds latency.
// Deterministic finish: per-wave partials -> d_ws -> tree-reduce kernel.
// ---------------------------------------------------------------------------

#define N_TOTAL 16384
#define MARGIN_F 0.1f
#define CHUNK 4096          // floats of pred + floats of target per chunk (32 KB)
#define JTILES (CHUNK / 16)
#define THREADS 256         // 8 wave32 per block
#define ROWTILES_PER_BLOCK 8
#define NUM_BLOCKS (N_TOTAL / (16 * ROWTILES_PER_BLOCK))   // 128
#define NUM_PARTIALS (NUM_BLOCKS * ROWTILES_PER_BLOCK * 2) // 2048 floats in d_ws

typedef float v2f __attribute__((ext_vector_type(2)));
typedef float v8f __attribute__((ext_vector_type(8)));

__global__ __launch_bounds__(THREADS)
void RankingLoss_34256659153340_kernel(const float* __restrict__ pred,
                                       const float* __restrict__ target,
                                       float* __restrict__ partials) {
    // [0, CHUNK): pred chunk; [CHUNK, 2*CHUNK): target chunk; +32 pad so the
    // one-ahead prefetch on the last tile stays inside the allocation.
    __shared__ float sbuf[2 * CHUNK + 32];

    const int tid  = threadIdx.x;
    const int lane = tid & 31;            // wave32
    const int wave = tid >> 5;
    const int m    = lane & 15;
    const bool hiHalf = lane >= 16;

    // This wave owns rows [i0, i0+16)
    const int i0 = (blockIdx.x * ROWTILES_PER_BLOCK + wave) * 16;

    // ---- A matrix (16x4 f32, constant for the wave) -----------------------
    // Lanes 0-15: K=0 (v0) / K=1 (v1), row M = lane -> (-p_i, 1).
    // Lanes 16-31: K=2 / K=3 -> zeroed, so B's K2/K3 rows (hi halves of b)
    // may carry arbitrary values and b needs no per-tile masking.
    const float pi = pred[i0 + m];
    v2f a;
    a[0] = hiHalf ? 0.0f : -pi;
    a[1] = hiHalf ? 0.0f : 1.0f;

    // t_i for the 8 rows each D-VGPR maps to (D: VGPR v -> row v / v+8 by half)
    float ti[8];
#pragma unroll
    for (int v = 0; v < 8; ++v)
        ti[v] = target[i0 + v + (hiHalf ? 8 : 0)];

    float rs[8];   // masked hinge partial sums (per column-class)
    float ct[8];   // masked counts, exact in f32 (<= 16384)
#pragma unroll
    for (int v = 0; v < 8; ++v) { rs[v] = 0.0f; ct[v] = 0.0f; }

    v8f cmat;
#pragma unroll
    for (int v = 0; v < 8; ++v) cmat[v] = MARGIN_F;

    for (int c = 0; c < N_TOTAL / CHUNK; ++c) {
        const int jbase = c * CHUNK;

        // Cooperative chunk load, float4-vectorized
        const float4* gp = (const float4*)(pred + jbase);
        const float4* gt = (const float4*)(target + jbase);
        float4* lp = (float4*)(sbuf);
        float4* lt = (float4*)(sbuf + CHUNK);
#pragma unroll
        for (int q = 0; q < CHUNK / 4 / THREADS; ++q) {
            lp[tid + q * THREADS] = gp[tid + q * THREADS];
            lt[tid + q * THREADS] = gt[tid + q * THREADS];
        }
        __syncthreads();

        // Software pipeline: (pj, tj) loaded one tile ahead.
        float pj = sbuf[m];
        float tj = sbuf[CHUNK + m];
#pragma unroll 2
        for (int jt = 0; jt < JTILES; ++jt) {
            const int nl = (jt + 1) * 16 + m;       // prefetch (pad-covered)
            const float pjn = sbuf[nl];
            const float tjn = sbuf[CHUNK + nl];

            // ---- B (4x16 f32): row 0 = ones, row 1 = p_j; K2/K3 rows are
            // don't-care because A's K2/K3 columns are zero.
            v2f b;
            b[0] = 1.0f;
            b[1] = pj;

            // D[i,j] = margin - p_i + p_j on the matrix pipe
            v8f d = __builtin_amdgcn_wmma_f32_16x16x4_f32(
                /*neg_a=*/false, a, /*neg_b=*/false, b,
                /*c_mod=*/(short)0, cmat, /*reuse_a=*/false, /*reuse_b=*/false);

#pragma unroll
            for (int v = 0; v < 8; ++v) {
                const bool take = tj < ti[v];          // 1 v_cmp, mask reused
                const float x = take ? d[v] : -1.0f;   // dual cndmask
                rs[v] += __builtin_amdgcn_fmed3f(x, 0.0f, 3.0e38f); // pk_add
                ct[v] += take ? 1.0f : 0.0f;           // dual cndmask + pk_add
            }

            pj = pjn;
            tj = tjn;
        }
        __syncthreads();
    }

    // Reduce the 16 column-classes: xor-shuffles stay within each 16-lane
    // half (rows 0-7 live in lanes 0-15, rows 8-15 in lanes 16-31).
#pragma unroll
    for (int v = 0; v < 8; ++v) {
#pragma unroll
        for (int msk = 1; msk < 16; msk <<= 1) {
            rs[v] += __shfl_xor(rs[v], msk, 32);
            ct[v] += __shfl_xor(ct[v], msk, 32);
        }
    }

    if (m == 0) {  // lanes 0 and 16 hold complete row sums for their half
        float local = 0.0f;
#pragma unroll
        for (int v = 0; v < 8; ++v)
            local += (ct[v] > 0.0f) ? (rs[v] / ct[v]) : 0.0f;
        const int waveGlobal = blockIdx.x * ROWTILES_PER_BLOCK + wave;
        partials[waveGlobal * 2 + (hiHalf ? 1 : 0)] = local;
    }
}

// Deterministic final reduction of the 2048 per-wave partials.
__global__ __launch_bounds__(THREADS)
void RankingLoss_reduce_kernel(const float* __restrict__ partials,
                               float* __restrict__ out) {
    __shared__ float s[THREADS];
    const int tid = threadIdx.x;
    float acc = 0.0f;
#pragma unroll
    for (int k = 0; k < NUM_PARTIALS / THREADS; ++k)
        acc += partials[tid + k * THREADS];
    s[tid] = acc;
    __syncthreads();
    for (int off = THREADS / 2; off > 0; off >>= 1) {
        if (tid < off) s[tid] += s[tid + off];
        __syncthreads();
    }
    if (tid == 0) out[0] = s[0] * (1.0f / (float)N_TOTAL);
}

extern "C" void kernel_launch(void* const* d_in, const int* in_sizes, int n_in,
                              void* d_out, int out_size, void* d_ws, size_t ws_size,
                              hipStream_t stream) {
    (void)in_sizes; (void)n_in; (void)out_size; (void)ws_size;
    const float* pred   = (const float*)d_in[0];
    const float* target = (const float*)d_in[1];
    float* out      = (float*)d_out;
    float* partials = (float*)d_ws;          // needs 8 KB of scratch

    dim3 grid(NUM_BLOCKS);                   // 128 blocks
    dim3 block(THREADS);                     // 8 wave32 each
    RankingLoss_34256659153340_kernel<<<grid, block, 0, stream>>>(pred, target, partials);
    RankingLoss_reduce_kernel<<<1, block, 0, stream>>>(partials, out);
}